// VQAModel_21010980012630
// MI455X (gfx1250) — compile-verified
//
#include <hip/hip_runtime.h>
#include <math.h>

// ---------------- model dimensions ----------------
#define BB    64        // batch
#define EE    256       // embedding dim
#define HH    512       // lstm hidden
#define H4    2048      // 4*H
#define DIMG  2048      // image feature dim
#define CC    1000      // num classes
#define PP    50000     // program vocab
#define TT    32        // program seq len
#define LQ    30        // question len
#define SOS   1

typedef __attribute__((ext_vector_type(2))) float v2f;
typedef __attribute__((ext_vector_type(8))) float v8f;

__device__ __forceinline__ float sigmoidf_(float x) {
  return 1.0f / (1.0f + __expf(-x));
}

// ============================================================
// Register-blocked fp32 WMMA GEMM:
//   C[M,N] = A[M,K] @ W[N,K]^T (+bias)(+relu)
// One wave per block. A wave owns MT m-tiles x NTW n-tiles, so each
// B fragment (weight tile) is reused MT times and each A fragment NTW
// times: loads/WMMA = (MT+NTW)/(MT*NTW) vs 2 unblocked. This is what
// keeps the 102MB out_W projection near its L2-traffic floor.
// Uses V_WMMA_F32_16X16X4_F32 (exact fp32 FMA, RNE) per CDNA5 ISA 7.12.
// A frag (16x4): lane r holds A[r, k0+2*half .. +1]  (half = lane>>4)
// B frag (4x16): lane r holds W[n0+r, k0+2*half .. +1]  (B[k,n] = W[n,k])
// C frag (16x16): vgpr vr, lane -> (m = vr + 8*half, n = r)
// ============================================================
template <int MT, int NTW, int ACT, bool NGUARD>
__global__ void gemm_wmma_f32(const float* __restrict__ A, int lda,
                              const float* __restrict__ W, int K,
                              const float* __restrict__ bias,
                              float* __restrict__ C, long long ldc,
                              int Nact) {
  const int lane = threadIdx.x & 31;
  const int half = lane >> 4;
  const int r    = lane & 15;
  const int m0   = blockIdx.x * (MT * 16);
  const int n0   = blockIdx.y * (NTW * 16);

  v8f acc[MT][NTW];
#pragma unroll
  for (int i = 0; i < MT; ++i)
#pragma unroll
    for (int j = 0; j < NTW; ++j) acc[i][j] = v8f{};

  for (int k0 = 0; k0 < K; k0 += 4) {
    v2f a[MT];
#pragma unroll
    for (int i = 0; i < MT; ++i) {
      const float* pa = A + (size_t)(m0 + i * 16 + r) * lda + k0 + 2 * half;
      a[i].x = pa[0]; a[i].y = pa[1];
    }
#pragma unroll
    for (int j = 0; j < NTW; ++j) {
      int wrow = n0 + j * 16 + r;
      if (NGUARD) wrow = (wrow < Nact) ? wrow : (Nact - 1);
      const float* pw = W + (size_t)wrow * K + k0 + 2 * half;
      v2f b; b.x = pw[0]; b.y = pw[1];
#pragma unroll
      for (int i = 0; i < MT; ++i)
        acc[i][j] = __builtin_amdgcn_wmma_f32_16x16x4_f32(
            false, a[i], false, b, (short)0, acc[i][j], false, false);
    }
  }

#pragma unroll
  for (int j = 0; j < NTW; ++j) {
    int n = n0 + j * 16 + r;
    float bv = 0.0f;
    if (bias) {
      int nb = n;
      if (NGUARD) nb = (nb < Nact) ? nb : (Nact - 1);
      bv = bias[nb];
    }
#pragma unroll
    for (int i = 0; i < MT; ++i) {
#pragma unroll
      for (int vr = 0; vr < 8; ++vr) {
        int m = m0 + i * 16 + vr + 8 * half;
        float v = acc[i][j][vr] + bv;
        if (ACT == 1) v = fmaxf(v, 0.0f);
        if (!NGUARD || n < Nact) C[(size_t)m * ldc + n] = v;
      }
    }
  }
}

// ============================================================
// Fused LSTM step: G = A @ W4^T (+pre) (+bih+bhh), gates i,f,g,o,
//   c = sig(f)*c + sig(i)*tanh(g);  h = sig(o)*tanh(c)
// W4 is [4H, K] (PyTorch layout, gate order i,f,g,o).
// Each wave owns tile (tm, tn) and accumulates the 4 gate tiles at
// columns tn*16 + {0,H,2H,3H}, so the nonlinearity runs directly on
// the WMMA accumulator fragments. grid=(H/16)=32 blocks, block=128 (4 waves).
// c is updated in place (each element owned by one lane); h -> hout.
// ============================================================
__global__ void lstm_step_wmma(const float* __restrict__ A, int lda, int K,
                               const float* __restrict__ W4,
                               const float* __restrict__ pre,   // [B,4H] or null
                               const float* __restrict__ bih,
                               const float* __restrict__ bhh,
                               float* __restrict__ c,            // [B,H]
                               float* __restrict__ hout, int ldh) {
  const int lane = threadIdx.x & 31;
  const int tm   = threadIdx.x >> 5;   // 0..3  (B=64 -> 4 m-tiles)
  const int tn   = blockIdx.x;         // 0..31 (H=512 -> 32 n-tiles)
  const int half = lane >> 4;
  const int r    = lane & 15;
  const int m0   = tm * 16;
  const int ncol = tn * 16 + r;        // column within one gate block [0,H)

  v8f acc[4];
#pragma unroll
  for (int g = 0; g < 4; ++g) acc[g] = v8f{};

  for (int k0 = 0; k0 < K; k0 += 4) {
    const float* pa = A + (size_t)(m0 + r) * lda + k0 + 2 * half;
    v2f a; a.x = pa[0]; a.y = pa[1];
#pragma unroll
    for (int g = 0; g < 4; ++g) {
      const int wrow = g * HH + tn * 16 + r;
      const float* pw = W4 + (size_t)wrow * K + k0 + 2 * half;
      v2f b; b.x = pw[0]; b.y = pw[1];
      acc[g] = __builtin_amdgcn_wmma_f32_16x16x4_f32(
          false, a, false, b, (short)0, acc[g], false, false);
    }
  }

  float bsum[4];
#pragma unroll
  for (int g = 0; g < 4; ++g) {
    const int j = g * HH + ncol;
    bsum[g] = bih[j] + bhh[j];
  }

#pragma unroll
  for (int vr = 0; vr < 8; ++vr) {
    const int m = m0 + vr + 8 * half;
    float xi = acc[0][vr] + bsum[0];
    float xf = acc[1][vr] + bsum[1];
    float xg = acc[2][vr] + bsum[2];
    float xo = acc[3][vr] + bsum[3];
    if (pre) {
      const float* pr = pre + (size_t)m * H4;
      xi += pr[0 * HH + ncol];
      xf += pr[1 * HH + ncol];
      xg += pr[2 * HH + ncol];
      xo += pr[3 * HH + ncol];
    }
    const size_t ci = (size_t)m * HH + ncol;
    float cn = sigmoidf_(xf) * c[ci] + sigmoidf_(xi) * tanhf(xg);
    c[ci] = cn;
    hout[(size_t)m * ldh + ncol] = sigmoidf_(xo) * tanhf(cn);
  }
}

// ---------------- small helper kernels ----------------

__global__ void zero_f(float* __restrict__ p, int n) {
  int i = blockIdx.x * blockDim.x + threadIdx.x;
  if (i < n) p[i] = 0.0f;
}

// emb_q[l*B + b, :] = embedding[questions[b*LQ + l]]   (grid=LQ*B, block=E)
__global__ void gather_emb_q(const float* __restrict__ emb,
                             const int* __restrict__ q,
                             float* __restrict__ out) {
  const int row = blockIdx.x;
  const int l = row >> 6;       // / 64
  const int b = row & 63;
  const int tok = q[b * LQ + l];
  out[(size_t)row * EE + threadIdx.x] = emb[(size_t)tok * EE + threadIdx.x];
}

// Wcat[j, 0:E] = Wih[j,:]; Wcat[j, E:E+H] = Whh[j,:]   (grid=4H, block=256)
__global__ void build_wcat(const float* __restrict__ Wih,
                           const float* __restrict__ Whh,
                           float* __restrict__ Wcat) {
  const int j = blockIdx.x;
  for (int k = threadIdx.x; k < EE + HH; k += blockDim.x)
    Wcat[(size_t)j * (EE + HH) + k] =
        (k < EE) ? Wih[(size_t)j * EE + k] : Whh[(size_t)j * HH + (k - EE)];
}

// xh[b, 0:E] = embedding[tok[b]] (or SOS if tok==null)   (grid=B, block=E)
__global__ void fill_x(const float* __restrict__ emb,
                       const int* __restrict__ tok,
                       float* __restrict__ xh, int ldx) {
  const int b = blockIdx.x;
  const int t = tok ? tok[b] : SOS;
  xh[(size_t)b * ldx + threadIdx.x] = emb[(size_t)t * EE + threadIdx.x];
}

// first-occurrence argmax over P per row (matches jnp.argmax). grid=B, block=256
__global__ void argmax_row(const float* __restrict__ logits, long long row_stride,
                           int P, int* __restrict__ tok) {
  __shared__ float sv[256];
  __shared__ int   si[256];
  const int b = blockIdx.x;
  const float* row = logits + (size_t)b * row_stride;
  float best = -INFINITY;
  int   bi   = P;
  for (int p = threadIdx.x; p < P; p += blockDim.x) {
    float v = row[p];
    if (v > best || (v == best && p < bi)) { best = v; bi = p; }
  }
  sv[threadIdx.x] = best; si[threadIdx.x] = bi;
  __syncthreads();
  for (int s = 128; s > 0; s >>= 1) {
    if (threadIdx.x < s) {
      float vo = sv[threadIdx.x + s]; int io = si[threadIdx.x + s];
      if (vo > sv[threadIdx.x] ||
          (vo == sv[threadIdx.x] && io < si[threadIdx.x])) {
        sv[threadIdx.x] = vo; si[threadIdx.x] = io;
      }
    }
    __syncthreads();
  }
  if (threadIdx.x == 0) tok[b] = si[0];
}

// ============================================================
extern "C" void kernel_launch(void* const* d_in, const int* in_sizes, int n_in,
                              void* d_out, int out_size, void* d_ws, size_t ws_size,
                              hipStream_t stream) {
  (void)in_sizes; (void)n_in; (void)out_size; (void)ws_size;

  const float* img_feat = (const float*)d_in[0];
  const int*   questions= (const int*)  d_in[1];
  const float* embedding= (const float*)d_in[2];
  const float* enc_Wih  = (const float*)d_in[3];
  const float* enc_Whh  = (const float*)d_in[4];
  const float* enc_bih  = (const float*)d_in[5];
  const float* enc_bhh  = (const float*)d_in[6];
  const float* img_W    = (const float*)d_in[7];
  const float* img_b    = (const float*)d_in[8];
  const float* cls_W    = (const float*)d_in[9];
  const float* cls_b    = (const float*)d_in[10];
  const float* dfc_W    = (const float*)d_in[11];
  const float* dfc_b    = (const float*)d_in[12];
  const float* dec_Wih  = (const float*)d_in[13];
  const float* dec_Whh  = (const float*)d_in[14];
  const float* dec_bih  = (const float*)d_in[15];
  const float* dec_bhh  = (const float*)d_in[16];
  const float* out_W    = (const float*)d_in[17];
  const float* out_b    = (const float*)d_in[18];
  float* out = (float*)d_out;

  // ---- workspace carve-out (fp32, 256B aligned chunks) ----
  float* ws = (float*)d_ws;
  size_t off = 0;
  auto alloc = [&](size_t n) { float* p = ws + off; off += (n + 63) & ~(size_t)63; return p; };
  float* emb_q  = alloc((size_t)LQ * BB * EE);        // [LQ*B, E]
  float* xw_pre = alloc((size_t)LQ * BB * H4);        // x@Wih^T for all enc steps
  float* hbuf0  = alloc((size_t)BB * 2 * HH);         // ping; doubles as `combined`
  float* hbuf1  = alloc((size_t)BB * 2 * HH);         // pong
  float* c_enc  = alloc((size_t)BB * HH);
  float* wcat   = alloc((size_t)H4 * (EE + HH));      // [dec_Wih | dec_Whh]
  float* xh0    = alloc((size_t)BB * (EE + HH));      // decoder [x|h] ping
  float* xh1    = alloc((size_t)BB * (EE + HH));      // pong
  float* c_dec  = alloc((size_t)BB * HH);
  int*   tok    = (int*)alloc(64);

  // ---- init state (must re-zero each call; harness does not) ----
  zero_f<<<dim3((BB * 2 * HH + 255) / 256), dim3(256), 0, stream>>>(hbuf0, BB * 2 * HH);
  zero_f<<<dim3((BB * HH + 255) / 256),     dim3(256), 0, stream>>>(c_enc, BB * HH);
  zero_f<<<dim3((BB * HH + 255) / 256),     dim3(256), 0, stream>>>(c_dec, BB * HH);

  // ---- encoder: gather embeddings, precompute x@Wih^T for all 30 steps ----
  gather_emb_q<<<dim3(LQ * BB), dim3(EE), 0, stream>>>(embedding, questions, emb_q);
  // [1920,2048] = emb_q[1920,256] @ enc_Wih^T ; 120 m-tiles (MT=2), 32 n-strips
  gemm_wmma_f32<2, 4, 0, false><<<dim3(60, 32), dim3(32), 0, stream>>>(
      emb_q, EE, enc_Wih, EE, nullptr, xw_pre, H4, H4);

  // ---- 30 sequential encoder LSTM steps (K=512, pre holds the x term) ----
  float* hin = hbuf0; float* hout_ = hbuf1;
  for (int t = 0; t < LQ; ++t) {
    lstm_step_wmma<<<dim3(HH / 16), dim3(128), 0, stream>>>(
        hin, 2 * HH, HH, enc_Whh, xw_pre + (size_t)t * BB * H4,
        enc_bih, enc_bhh, c_enc, hout_, 2 * HH);
    float* tmp = hin; hin = hout_; hout_ = tmp;
  }
  // 30 steps (even) -> final h_n lives in hbuf0[:, 0:H];  combined = hbuf0

  // ---- image branch into combined[:, H:2H] with ReLU (MT=4: W read once) ----
  gemm_wmma_f32<4, 4, 1, false><<<dim3(1, 8), dim3(32), 0, stream>>>(
      img_feat, DIMG, img_W, DIMG, img_b, hbuf0 + HH, 2 * HH, HH);

  // ---- classifier -> d_out[0 : B*C]   (N=1000, guarded tiles) ----
  gemm_wmma_f32<1, 1, 0, true><<<dim3(4, 63), dim3(32), 0, stream>>>(
      hbuf0, 2 * HH, cls_W, 2 * HH, cls_b, out, CC, CC);

  // ---- decoder init: h0 = relu(combined@dfc^T+b) -> xh0[:, E:E+H] ----
  gemm_wmma_f32<4, 4, 1, false><<<dim3(1, 8), dim3(32), 0, stream>>>(
      hbuf0, 2 * HH, dfc_W, 2 * HH, dfc_b, xh0 + EE, EE + HH, HH);
  build_wcat<<<dim3(H4), dim3(256), 0, stream>>>(dec_Wih, dec_Whh, wcat);
  fill_x<<<dim3(BB), dim3(EE), 0, stream>>>(embedding, nullptr, xh0, EE + HH);

  // ---- 32 greedy decoder steps ----
  float* prog = out + (size_t)BB * CC;           // program_outputs [B,T,P]
  const long long ldl = (long long)TT * PP;      // row stride over b
  for (int t = 0; t < TT; ++t) {
    float* in  = (t & 1) ? xh1 : xh0;
    float* nxt = (t & 1) ? xh0 : xh1;
    // LSTM cell on [x|h], K=768; h written into next step's buffer
    lstm_step_wmma<<<dim3(HH / 16), dim3(128), 0, stream>>>(
        in, EE + HH, EE + HH, wcat, nullptr,
        dec_bih, dec_bhh, c_dec, nxt + EE, EE + HH);
    // logits[64,50000] = h @ out_W^T + out_b  -> d_out at [b, t, :]
    // MT=2 x NTW=5: out_W tile traffic halved, 1250 waves of work
    gemm_wmma_f32<2, 5, 0, false><<<dim3(2, 625), dim3(32), 0, stream>>>(
        nxt + EE, EE + HH, out_W, HH, out_b, prog + (size_t)t * PP, ldl, PP);
    if (t < TT - 1) {
      argmax_row<<<dim3(BB), dim3(256), 0, stream>>>(
          prog + (size_t)t * PP, ldl, PP, tok);
      fill_x<<<dim3(BB), dim3(EE), 0, stream>>>(embedding, tok, nxt, EE + HH);
    }
  }
}